// CrossEntropyPAUCLossV2_42305427866234
// MI455X (gfx1250) — compile-verified
//
#include <hip/hip_runtime.h>
#include <hip/hip_bf16.h>

typedef float v2f __attribute__((ext_vector_type(2)));
typedef float v8f __attribute__((ext_vector_type(8)));
typedef float f4  __attribute__((ext_vector_type(4)));
typedef int   i4  __attribute__((ext_vector_type(4)));

#define BLOCK1 256
#define GRID1 2048          // ce_partials count; k2b assumes GRID1 == 2*T2
#define T2 1024
#define RECALL_LO_C 0.95f

__device__ __forceinline__ void ce_hist_elem(float l, int t, float& acc,
                                             unsigned int* __restrict__ hist_pos,
                                             unsigned int* __restrict__ hist_neg,
                                             int shift)
{
    // binary CE with logits: max(l,0) - l*t + log1p(exp(-|l|))
    acc += fmaxf(l, 0.0f) - l * (float)t + log1pf(expf(-fabsf(l)));
    // monotone ascending key of f32; descending order handled at scan time
    unsigned int u = __float_as_uint(l);
    unsigned int k = (u & 0x80000000u) ? ~u : (u | 0x80000000u);
    atomicAdd((t ? hist_pos : hist_neg) + (k >> shift), 1u);
}

// ---------------- Kernel 1: fused CE partial + descending-score histogram ----
// Streaming pass: 67 MB @ 23.3 TB/s ~ 2.9 us. b128 NT loads keep the read-once
// stream out of L2; the 8 MB histogram stays L2-resident for the atomics.
__global__ void k1_ce_hist(const float* __restrict__ pred,
                           const int* __restrict__ tgt, int n,
                           float* __restrict__ ce_partials,
                           unsigned int* __restrict__ hist_pos,
                           unsigned int* __restrict__ hist_neg,
                           int shift)
{
    const int tid = threadIdx.x;
    const int stride = GRID1 * BLOCK1;
    float acc = 0.0f;

    const int nvec = n >> 2;
    const f4* __restrict__ pred4 = (const f4*)pred;
    const i4* __restrict__ tgt4  = (const i4*)tgt;
    for (int i = blockIdx.x * BLOCK1 + tid; i < nvec; i += stride) {
        if (i + stride < nvec) {
            __builtin_prefetch(&pred4[i + stride], 0, 1);   // global_prefetch_b8
            __builtin_prefetch(&tgt4 [i + stride], 0, 1);
        }
        f4 l4 = __builtin_nontemporal_load(&pred4[i]);      // global_load_b128 (NT)
        i4 t4 = __builtin_nontemporal_load(&tgt4[i]);
        #pragma unroll
        for (int c = 0; c < 4; ++c)
            ce_hist_elem(l4[c], t4[c], acc, hist_pos, hist_neg, shift);
    }
    // tail for n not divisible by 4 (none for N = 2^23, kept for generality)
    for (int i = (nvec << 2) + blockIdx.x * BLOCK1 + tid; i < n; i += stride)
        ce_hist_elem(pred[i], tgt[i], acc, hist_pos, hist_neg, shift);

    // Exact f32 wave-sum of `acc` via V_WMMA_F32_16X16X4_F32:
    // A = ones(16x4), B VGPR0 = acc (one f32/lane), B VGPR1 = 0.
    // D[m][n] = colsum_n(B) for every m; sum of d[0] over lanes 0..15
    // equals the total of all B entries = wave sum, independent of B's
    // exact slot bijection.
    v2f a; a[0] = 1.0f; a[1] = 1.0f;
    v2f b; b[0] = acc;  b[1] = 0.0f;
    v8f c = {};
    v8f d = __builtin_amdgcn_wmma_f32_16x16x4_f32(false, a, false, b,
                                                  (short)0, c, false, false);
    __shared__ float sdata[BLOCK1];
    sdata[tid] = d[0];
    __syncthreads();
    if (tid == 0) {
        float s = 0.0f;
        #pragma unroll
        for (int w = 0; w < BLOCK1 / 32; ++w)
            #pragma unroll
            for (int j = 0; j < 16; ++j)
                s += sdata[w * 32 + j];
        ce_partials[blockIdx.x] = s;   // deterministic per-block slot
    }
}

// ---------------- Kernel 2a: per-chunk pos/neg totals (descending order) -----
__global__ void k2a_chunk_sums(const unsigned int* __restrict__ hist_pos,
                               const unsigned int* __restrict__ hist_neg,
                               int nbins,
                               float* __restrict__ chunk_pos,
                               float* __restrict__ chunk_neg)
{
    __shared__ float s0[T2], s1[T2];
    int t = threadIdx.x;
    int q = blockIdx.x * T2 + t;       // q ascending == score descending
    int b = nbins - 1 - q;
    s0[t] = (float)hist_pos[b];
    s1[t] = (float)hist_neg[b];
    __syncthreads();
    for (int off = T2 / 2; off > 0; off >>= 1) {
        if (t < off) { s0[t] += s0[t + off]; s1[t] += s1[t + off]; }
        __syncthreads();
    }
    if (t == 0) { chunk_pos[blockIdx.x] = s0[0]; chunk_neg[blockIdx.x] = s1[0]; }
}

// ---------------- Kernel 2b: carries (exclusive scan), P, Ng, CE total -------
__global__ void k2b_carries(const float* __restrict__ chunk_pos,
                            const float* __restrict__ chunk_neg,
                            int nchunks,
                            const float* __restrict__ ce_partials,
                            float* __restrict__ carry,
                            float* __restrict__ scalars)
{
    __shared__ float s0[T2], s1[T2];
    int t = threadIdx.x;
    float v = (t < nchunks) ? chunk_pos[t] : 0.0f;
    s0[t] = v;
    __syncthreads();
    for (int off = 1; off < T2; off <<= 1) {   // Hillis-Steele inclusive scan
        float add = (t >= off) ? s0[t - off] : 0.0f;
        __syncthreads();
        s0[t] += add;
        __syncthreads();
    }
    float excl = s0[t] - v;
    if (t < nchunks) carry[t] = excl;
    float Ptot = s0[T2 - 1];
    // Ng total
    s1[t] = (t < nchunks) ? chunk_neg[t] : 0.0f;
    __syncthreads();
    for (int off = T2 / 2; off > 0; off >>= 1) {
        if (t < off) s1[t] += s1[t + off];
        __syncthreads();
    }
    float Ngtot = s1[0];
    __syncthreads();
    // CE total: GRID1 == 2*T2 partials
    s0[t] = ce_partials[t] + ce_partials[t + T2];
    __syncthreads();
    for (int off = T2 / 2; off > 0; off >>= 1) {
        if (t < off) s0[t] += s0[t + off];
        __syncthreads();
    }
    if (t == 0) { scalars[0] = Ptot; scalars[1] = Ngtot; scalars[2] = s0[0]; }
}

// ---------------- Kernel 2c: per-chunk pAUC contributions --------------------
// Each bin is a tie-run; its negatives contribute trapezoid of
// y = clip(TPR-0.95, 0) at the run's entry/exit cumulative-positive counts.
__global__ void k2c_contrib(const unsigned int* __restrict__ hist_pos,
                            const unsigned int* __restrict__ hist_neg,
                            int nbins,
                            const float* __restrict__ carry,
                            const float* __restrict__ scalars,
                            float* __restrict__ contrib)
{
    __shared__ float s0[T2];
    int t = threadIdx.x;
    int j = blockIdx.x;
    int q = j * T2 + t;
    int b = nbins - 1 - q;
    float p  = (float)hist_pos[b];
    float nn = (float)hist_neg[b];
    s0[t] = p;
    __syncthreads();
    for (int off = 1; off < T2; off <<= 1) {   // inclusive scan of positives
        float add = (t >= off) ? s0[t - off] : 0.0f;
        __syncthreads();
        s0[t] += add;
        __syncthreads();
    }
    float Pf  = scalars[0];
    float cpi = carry[j] + s0[t];   // positives through this bin (inclusive)
    float cpe = cpi - p;            // positives through previous bin
    float tpr_hi = (Pf > 0.0f) ? cpi / Pf : 0.0f;
    float tpr_lo = (Pf > 0.0f) ? cpe / Pf : 0.0f;
    float y_hi = fmaxf(tpr_hi - RECALL_LO_C, 0.0f);
    float y_lo = fmaxf(tpr_lo - RECALL_LO_C, 0.0f);
    float cc = nn * 0.5f * (y_lo + y_hi);
    __syncthreads();
    s0[t] = cc;
    __syncthreads();
    for (int off = T2 / 2; off > 0; off >>= 1) {
        if (t < off) s0[t] += s0[t + off];
        __syncthreads();
    }
    if (t == 0) contrib[j] = s0[0];
}

// ---------------- Kernel 2d: final combine -----------------------------------
__global__ void k2d_final(const float* __restrict__ contrib, int nchunks,
                          const float* __restrict__ scalars,
                          float* __restrict__ out, int n)
{
    __shared__ float s0[T2];
    int t = threadIdx.x;
    s0[t] = (t < nchunks) ? contrib[t] : 0.0f;
    __syncthreads();
    for (int off = T2 / 2; off > 0; off >>= 1) {
        if (t < off) s0[t] += s0[t + off];
        __syncthreads();
    }
    if (t == 0) {
        float Ng   = scalars[1];
        float pauc = (Ng > 0.0f) ? s0[0] / Ng : 0.0f;
        float avg  = pauc / (2.0f * (1.0f - RECALL_LO_C));
        avg = fminf(fmaxf(avg, 0.0f), 1.0f);
        float pauc_loss = 1.0f - avg * avg;
        float ce = scalars[2] / (float)n;
        out[0] = 0.5f * ce + 0.5f * pauc_loss;   // LAMBDA_PAUC = 0.5
    }
}

extern "C" void kernel_launch(void* const* d_in, const int* in_sizes, int n_in,
                              void* d_out, int out_size, void* d_ws, size_t ws_size,
                              hipStream_t stream)
{
    const float* pred = (const float*)d_in[0];
    const int*   tgt  = (const int*)d_in[1];
    int n = in_sizes[1];

    char* w = (char*)d_ws;
    float* scalars     = (float*)(w + 0);                 // [P, Ng, ce_sum]
    float* ce_partials = (float*)(w + 64);                // GRID1 floats
    float* chunk_pos   = (float*)(w + 64 + GRID1 * 4);    // T2 floats
    float* chunk_neg   = chunk_pos + T2;
    float* carry       = chunk_neg + T2;
    float* contrib     = carry + T2;

    size_t hist_off = (size_t)64 + GRID1 * 4 + (size_t)4 * T2 * 4;
    hist_off = (hist_off + 255) & ~(size_t)255;

    int log2bins = 20;                                    // 1M bins, 8 MB (L2-resident)
    while (log2bins > 12 && hist_off + ((size_t)8 << log2bins) > ws_size) --log2bins;
    int nbins   = 1 << log2bins;
    int shift   = 32 - log2bins;
    int nchunks = nbins / T2;                             // <= 1024

    unsigned int* hist_pos = (unsigned int*)(w + hist_off);
    unsigned int* hist_neg = hist_pos + nbins;

    hipMemsetAsync(w + hist_off, 0, (size_t)8 * nbins, stream);
    k1_ce_hist    <<<GRID1,   BLOCK1, 0, stream>>>(pred, tgt, n, ce_partials,
                                                   hist_pos, hist_neg, shift);
    k2a_chunk_sums<<<nchunks, T2,     0, stream>>>(hist_pos, hist_neg, nbins,
                                                   chunk_pos, chunk_neg);
    k2b_carries   <<<1,       T2,     0, stream>>>(chunk_pos, chunk_neg, nchunks,
                                                   ce_partials, carry, scalars);
    k2c_contrib   <<<nchunks, T2,     0, stream>>>(hist_pos, hist_neg, nbins,
                                                   carry, scalars, contrib);
    k2d_final     <<<1,       T2,     0, stream>>>(contrib, nchunks, scalars,
                                                   (float*)d_out, n);
}